// PartitionGNN_48533130445249
// MI455X (gfx1250) — compile-verified
//
#include <hip/hip_runtime.h>
#include <hip/hip_bf16.h>

// ---------------------------------------------------------------------------
// GCN (2-layer, sym-norm, self-loops) for MI455X / gfx1250 (wave32, WMMA).
//
// Pipeline:
//   deg  = 1 + scatter-count(dst)          (fp32 atomics)
//   dinv = rsqrt(deg)
//   L1:  h = WMMA_GEMM(x, W1)  -> agg = selfloop + edge-scatter -> relu(+b1)
//   L2:  h = WMMA_GEMM(h, W2)  -> agg = selfloop + edge-scatter -> relu(+b2)
//   out  = wave-reduced dot(h, Wout) + bout
// ---------------------------------------------------------------------------

#define N_NODES 100000
#define N_EDGES 3200000
#define F_IN    16
#define HID     32

typedef __attribute__((ext_vector_type(2))) float v2f;
typedef __attribute__((ext_vector_type(8))) float v8f;

// ---------------------------------------------------------------------------
// Degree / norm kernels
// ---------------------------------------------------------------------------
__global__ void deg_init_kernel(float* __restrict__ deg, int n) {
  int i = blockIdx.x * blockDim.x + threadIdx.x;
  if (i < n) deg[i] = 1.0f;  // self-loop contributes 1 to every node's degree
}

__global__ void deg_count_kernel(const int* __restrict__ dst,
                                 float* __restrict__ deg, int ne) {
  int e = blockIdx.x * blockDim.x + threadIdx.x;
  if (e < ne) atomicAdd(&deg[dst[e]], 1.0f);
}

__global__ void dinv_kernel(float* __restrict__ deg, int n) {
  int i = blockIdx.x * blockDim.x + threadIdx.x;
  if (i < n) {
    float d = deg[i];
    deg[i] = (d > 0.0f) ? rsqrtf(d) : 0.0f;  // in-place: deg -> dinv
  }
}

// ---------------------------------------------------------------------------
// Dense GEMM via V_WMMA_F32_16X16X4_F32.
// One wave computes a 16-row x 32-col output tile. K is stepped by 4.
//
// Fragment layouts (ISA 7.12.2, wave32):
//   A (16x4 f32):  lanes 0-15 -> M=lane,   VGPR0=K0, VGPR1=K1
//                  lanes 16-31-> M=lane-16,VGPR0=K2, VGPR1=K3
//   B (4x16 f32):  lanes 0-15 -> N=lane,   VGPR0=K0, VGPR1=K1
//                  lanes 16-31-> N=lane-16,VGPR0=K2, VGPR1=K3
//   C/D (16x16):   VGPR r -> M=r (lanes 0-15), M=r+8 (lanes 16-31); N=lane%16
// ---------------------------------------------------------------------------
template <int K>
__global__ void gemm_wmma_kernel(const float* __restrict__ X,   // [nrows, K]
                                 const float* __restrict__ W,   // [K, 32]
                                 float* __restrict__ H,         // [nrows, 32]
                                 int nrows) {
  const int wave = (int)((blockIdx.x * blockDim.x + threadIdx.x) >> 5);
  const int lane = (int)(threadIdx.x & 31);
  const int row0 = wave * 16;
  if (row0 >= nrows) return;  // wave-uniform: EXEC stays all-1s for WMMA

  const int half = lane >> 4;   // 0: lanes 0-15, 1: lanes 16-31
  const int m    = lane & 15;   // M (for A) / N (for B, C, D)

  v8f c0 = {};  // output cols 0..15
  v8f c1 = {};  // output cols 16..31

  const float* __restrict__ xrow = X + (size_t)(row0 + m) * K;

#pragma unroll
  for (int k0 = 0; k0 < K; k0 += 4) {
    const int kr = k0 + 2 * half;  // this lane holds K rows {kr, kr+1}
    v2f a;
    a.x = xrow[kr];
    a.y = xrow[kr + 1];
    v2f b0, b1;
    b0.x = W[(size_t)kr * HID + m];
    b0.y = W[(size_t)(kr + 1) * HID + m];
    b1.x = W[(size_t)kr * HID + 16 + m];
    b1.y = W[(size_t)(kr + 1) * HID + 16 + m];
    c0 = __builtin_amdgcn_wmma_f32_16x16x4_f32(false, a, false, b0,
                                               (short)0, c0, false, false);
    c1 = __builtin_amdgcn_wmma_f32_16x16x4_f32(false, a, false, b1,
                                               (short)0, c1, false, false);
  }

  float* __restrict__ hbase = H + (size_t)row0 * HID;
#pragma unroll
  for (int r = 0; r < 8; ++r) {
    const int mm = r + 8 * half;
    hbase[(size_t)mm * HID + m]      = c0[r];
    hbase[(size_t)mm * HID + 16 + m] = c1[r];
  }
}

// ---------------------------------------------------------------------------
// Self-loop init: agg[i,f] = h[i,f] * dinv[i]^2     (norm of the i->i edge)
// One wave per node, one feature per lane (fully coalesced).
// ---------------------------------------------------------------------------
__global__ void selfloop_init_kernel(const float* __restrict__ dinv,
                                     const float* __restrict__ h,
                                     float* __restrict__ agg, int n) {
  const long long t = (long long)blockIdx.x * blockDim.x + threadIdx.x;
  const int i = (int)(t >> 5);
  const int f = (int)(t & 31);
  if (i >= n) return;
  const float di = dinv[i];
  agg[(size_t)i * HID + f] = h[(size_t)i * HID + f] * (di * di);
}

// ---------------------------------------------------------------------------
// Edge scatter: one edge per wave, one feature per lane.
//   agg[dst, f] += h[src, f] * dinv[src] * dinv[dst]
// Coalesced gather + coalesced L2 atomics (agg fits entirely in 192MB L2).
// ---------------------------------------------------------------------------
__global__ void scatter_edges_kernel(const int* __restrict__ src,
                                     const int* __restrict__ dst,
                                     const float* __restrict__ dinv,
                                     const float* __restrict__ h,
                                     float* __restrict__ agg, int ne) {
  const long long t = (long long)blockIdx.x * blockDim.x + threadIdx.x;
  const int e = (int)(t >> 5);
  const int f = (int)(t & 31);
  if (e >= ne) return;
  const int s = src[e];
  const int d = dst[e];
  const float w = dinv[s] * dinv[d];
  const float v = h[(size_t)s * HID + f] * w;
  atomicAdd(&agg[(size_t)d * HID + f], v);
}

// ---------------------------------------------------------------------------
// Bias + ReLU, in place.
// ---------------------------------------------------------------------------
__global__ void bias_relu_kernel(float* __restrict__ a,
                                 const float* __restrict__ b, long long total) {
  const long long t = (long long)blockIdx.x * blockDim.x + threadIdx.x;
  if (t >= total) return;
  const int f = (int)(t & 31);
  const float v = a[t] + b[f];
  a[t] = v > 0.0f ? v : 0.0f;
}

// ---------------------------------------------------------------------------
// Output layer: out[i] = dot(h[i,:], Wout) + bout.  One node per wave,
// wave32 xor-shuffle reduction.
// ---------------------------------------------------------------------------
__global__ void out_kernel(const float* __restrict__ h,
                           const float* __restrict__ Wout,
                           const float* __restrict__ bout,
                           float* __restrict__ out, int n) {
  const long long t = (long long)blockIdx.x * blockDim.x + threadIdx.x;
  const int i = (int)(t >> 5);
  const int f = (int)(t & 31);
  if (i >= n) return;
  float v = h[(size_t)i * HID + f] * Wout[f];
#pragma unroll
  for (int off = 16; off > 0; off >>= 1) v += __shfl_xor(v, off, 32);
  if (f == 0) out[i] = v + bout[0];
}

// ---------------------------------------------------------------------------
// Launcher
// ---------------------------------------------------------------------------
extern "C" void kernel_launch(void* const* d_in, const int* in_sizes, int n_in,
                              void* d_out, int out_size, void* d_ws,
                              size_t ws_size, hipStream_t stream) {
  const float* x    = (const float*)d_in[0];      // [N, 16]
  const int*   ei   = (const int*)d_in[1];        // [2, E] (row0=src, row1=dst)
  const float* W1   = (const float*)d_in[2];      // [16, 32]
  const float* b1   = (const float*)d_in[3];      // [32]
  const float* W2   = (const float*)d_in[4];      // [32, 32]
  const float* b2   = (const float*)d_in[5];      // [32]
  const float* Wout = (const float*)d_in[6];      // [32, 1]
  const float* bout = (const float*)d_in[7];      // [1]
  float* out = (float*)d_out;                     // [N, 1]

  const int n  = N_NODES;
  const int ne = N_EDGES;
  const int* src = ei;
  const int* dst = ei + ne;

  // Workspace: dinv[N] | buf1[N*32] | buf2[N*32]   (~26 MB)
  float* dinv = (float*)d_ws;
  float* buf1 = dinv + n;
  float* buf2 = buf1 + (size_t)n * HID;

  const int TB = 256;
  const int gN      = (n + TB - 1) / TB;                        // node-parallel
  const int gE      = (ne + TB - 1) / TB;                       // edge-parallel
  const int gN32    = (int)(((long long)n * 32 + TB - 1) / TB); // node x feat
  const int gE32    = (int)(((long long)ne * 32 + TB - 1) / TB);// edge x feat
  const int nTiles  = (n + 15) / 16;                            // 16-row tiles
  const int gGemm   = (nTiles * 32 + TB - 1) / TB;              // 1 wave/tile

  // --- degree + symmetric norm -------------------------------------------
  deg_init_kernel<<<gN, TB, 0, stream>>>(dinv, n);
  deg_count_kernel<<<gE, TB, 0, stream>>>(dst, dinv, ne);
  dinv_kernel<<<gN, TB, 0, stream>>>(dinv, n);

  // --- layer 1: h = relu(scatter(x @ W1) + b1) ---------------------------
  gemm_wmma_kernel<F_IN><<<gGemm, TB, 0, stream>>>(x, W1, buf1, n);
  selfloop_init_kernel<<<gN32, TB, 0, stream>>>(dinv, buf1, buf2, n);
  scatter_edges_kernel<<<gE32, TB, 0, stream>>>(src, dst, dinv, buf1, buf2, ne);
  bias_relu_kernel<<<gN32, TB, 0, stream>>>(buf2, b1, (long long)n * HID);

  // --- layer 2: h = relu(scatter(h @ W2) + b2) ---------------------------
  gemm_wmma_kernel<HID><<<gGemm, TB, 0, stream>>>(buf2, W2, buf1, n);
  selfloop_init_kernel<<<gN32, TB, 0, stream>>>(dinv, buf1, buf2, n);
  scatter_edges_kernel<<<gE32, TB, 0, stream>>>(src, dst, dinv, buf1, buf2, ne);
  bias_relu_kernel<<<gN32, TB, 0, stream>>>(buf2, b2, (long long)n * HID);

  // --- readout -----------------------------------------------------------
  out_kernel<<<gN32, TB, 0, stream>>>(buf2, Wout, bout, out, n);
}